// GroupDelay_36661840839523
// MI455X (gfx1250) — compile-verified
//
#include <hip/hip_runtime.h>
#include <hip/hip_bf16.h>

// ---------------- problem constants ----------------
#define NB     32          // batch
#define NSAMP  160000      // samples per batch
#define WIN    400
#define HOP    160
#define NF     998         // frames per batch: 1 + (160000-400)/160
#define ORDER  16
#define NFREQ  257         // 512/2 + 1
#define KPAD   260         // NFREQ padded to multiple of 4 for WMMA K
#define NMEL   80
#define ROWS   (NB * NF)   // 31936 total frames = 1996 * 16
#define MTILES 1996        // ROWS / 16
#define NTILES 5           // NMEL / 16
#define WPB    4           // waves (mTiles) per block in the WMMA kernel

typedef __attribute__((ext_vector_type(2))) float v2f;
typedef __attribute__((ext_vector_type(8))) float v8f;

// =====================================================================
// Kernel 0: pack mel_fb (80 x 257, row-major) into a transposed,
// K-padded panel melT[KPAD][NMEL] with zeros for k >= 257, so the WMMA
// kernel's B loads are completely unconditional.
// =====================================================================
__global__ void mel_pack(const float* __restrict__ mel_fb,
                         float* __restrict__ melT) {
    const int idx = blockIdx.x * 256 + threadIdx.x;   // over KPAD*NMEL
    if (idx >= KPAD * NMEL) return;
    const int k = idx / NMEL;
    const int n = idx % NMEL;
    melT[idx] = (k < NFREQ) ? mel_fb[n * NFREQ + k] : 0.0f;
}

// =====================================================================
// Kernel 1: pre-emphasis + window + 17-lag autocorrelation + Levinson.
// One wave (32 lanes) per frame; 8 waves (256 threads) per block.
// (Direct time-domain autocorr == the reference's rfft/irfft path since
//  512 >= 400+16, so circular == linear for lags <= 16.)
// =====================================================================
__global__ void gd_autocorr_levinson(const float* __restrict__ feat,
                                     const float* __restrict__ window,
                                     float* __restrict__ aWS) {
    __shared__ float sw[8][WIN + 16];   // zero-padded windowed frame per wave

    const int lane = threadIdx.x & 31;
    const int wave = threadIdx.x >> 5;
    const int f    = blockIdx.x * 8 + wave;       // frame index (may be >= NF)
    const int fc   = f < NF ? f : (NF - 1);       // clamp for safe redundant work
    const int b    = blockIdx.y;
    const long base = (long)b * NSAMP;
    const int g    = fc * HOP;                    // frame start sample

    for (int n = lane; n < WIN; n += 32) {
        const int gi = g + n;
        float xv = feat[base + gi];
        if (gi > 0) xv -= 0.97f * feat[base + gi - 1];
        sw[wave][n] = xv * window[n];
    }
    if (lane < 16) sw[wave][WIN + lane] = 0.0f;   // zero pad for lags
    __syncthreads();

    float part[ORDER + 1];
#pragma unroll
    for (int l = 0; l <= ORDER; ++l) part[l] = 0.0f;
    for (int n = lane; n < WIN; n += 32) {
        const float pn = sw[wave][n];
#pragma unroll
        for (int l = 0; l <= ORDER; ++l) part[l] = fmaf(pn, sw[wave][n + l], part[l]);
    }
#pragma unroll
    for (int off = 16; off > 0; off >>= 1) {
#pragma unroll
        for (int l = 0; l <= ORDER; ++l)
            part[l] += __shfl_xor(part[l], off, 32);
    }

    if (lane == 0 && f < NF) {
        float A[ORDER + 1];
        float T[ORDER + 1];
        A[0] = 1.0f;
        float e = part[0];
        for (int i = 1; i <= ORDER; ++i) {
            float acc = 0.0f;
            for (int j = 0; j < i; ++j) acc = fmaf(A[j], part[i - j], acc);
            const float k = -acc / (e + 1e-10f);
            for (int j = 1; j < i; ++j) T[j] = fmaf(k, A[i - j], A[j]);
            for (int j = 1; j < i; ++j) A[j] = T[j];
            A[i] = k;
            e *= (1.0f - k * k);
        }
        const long row = (long)b * NF + f;
#pragma unroll
        for (int j = 0; j <= ORDER; ++j) aWS[row * 17 + j] = A[j];
    }
}

// =====================================================================
// Kernel 2: Hr/Hi via Chebyshev cos/sin recurrence, atan2 phase,
// gd[k] = phase[k] - phase[k+1] (last bin duplicated). One block per
// frame; gd stored with stride KPAD=260, bins 257..259 zeroed so the
// WMMA kernel needs no A-side K guard.
// =====================================================================
__global__ void gd_phase(const float* __restrict__ aWS,
                         float* __restrict__ gdWS) {
    __shared__ float a_sh[ORDER + 1];
    __shared__ float ph[NFREQ + 3];

    const int tid = threadIdx.x;
    const long row = (long)blockIdx.y * NF + blockIdx.x;

    if (tid <= ORDER) a_sh[tid] = aWS[row * 17 + tid];
    __syncthreads();

    for (int k = tid; k < NFREQ; k += 256) {
        const float w    = 3.14159265358979323846f * (float)k * (1.0f / 256.0f);
        const float cosw = __cosf(w);
        const float sinw = __sinf(w);
        float Hr = a_sh[0];
        float Hi = 0.0f;
        float cp = 1.0f, cc = cosw;
        float sp = 0.0f, sc = sinw;
        const float twoc = 2.0f * cosw;
#pragma unroll
        for (int n = 1; n <= ORDER; ++n) {
            Hr = fmaf(a_sh[n], cc, Hr);
            Hi = fmaf(-a_sh[n], sc, Hi);
            const float cn = fmaf(twoc, cc, -cp); cp = cc; cc = cn;
            const float sn = fmaf(twoc, sc, -sp); sp = sc; sc = sn;
        }
        ph[k] = atan2f(Hi, Hr);
    }
    __syncthreads();

    for (int k = tid; k < NFREQ; k += 256) {
        float gd;
        if (k < NFREQ - 1)  gd = ph[k] - ph[k + 1];
        else                gd = ph[NFREQ - 2] - ph[NFREQ - 1];
        gdWS[row * KPAD + k] = gd;
    }
    if (tid < KPAD - NFREQ) gdWS[row * KPAD + NFREQ + tid] = 0.0f;
}

// =====================================================================
// Kernel 3: logmel = log(gd @ mel_fb^T + 1e-8) via V_WMMA_F32_16X16X4_F32.
// Block = 128 threads = 4 waves; each wave computes one 16x16 tile
// (its own mTile), all 4 share one nTile's B panel staged in LDS.
//   A frag (16x4 f32): lanes 0-15 -> K=k0,k0+1; lanes 16-31 -> K=k0+2,k0+3
//   B frag (4x16 f32): mirrored K split, N = lane%16
//   C/D (16x16 f32):   VGPR r -> M=r (lanes 0-15) / M=r+8 (lanes 16-31)
// LDS bsh[16][260]: lane bank = (lr*260 + ka) % 64 = 4*lr + ka(+2)
// -> the 32 b64 reads touch all 64 banks exactly once (conflict-free).
// =====================================================================
__global__ void gd_mel_wmma(const float* __restrict__ gdWS,
                            const float* __restrict__ melT,
                            float* __restrict__ out) {
    __shared__ float bsh[16][KPAD];   // B panel: [n within tile][k]

    const int tid  = threadIdx.x;     // 0..127
    const int lane = tid & 31;
    const int wave = tid >> 5;        // 0..3
    const int half = lane >> 4;
    const int lr   = lane & 15;

    const int nTile = blockIdx.y;     // 0..4
    const int n0    = nTile * 16;

    // cooperative B-panel stage: each thread handles whole k-columns,
    // reading 16 consecutive floats of melT row k as 4x float4
    for (int k = tid; k < KPAD; k += 128) {
        const float4* src = (const float4*)(melT + (long)k * NMEL + n0);
#pragma unroll
        for (int v = 0; v < 4; ++v) {
            const float4 q = src[v];
            bsh[4 * v + 0][k] = q.x;
            bsh[4 * v + 1][k] = q.y;
            bsh[4 * v + 2][k] = q.z;
            bsh[4 * v + 3][k] = q.w;
        }
    }
    __syncthreads();

    const int  mTile = blockIdx.x * WPB + wave;       // 0..1995
    const long aBase = ((long)mTile * 16 + lr) * KPAD;
    const int  nCol  = n0 + lr;

    v8f c = {};
    for (int k0 = 0; k0 < KPAD; k0 += 4) {
        const int ka = k0 + 2 * half;                 // this lane's K pair
        v2f a, bfr;
        a.x   = gdWS[aBase + ka];                     // -> global_load_b64
        a.y   = gdWS[aBase + ka + 1];
        bfr.x = bsh[lr][ka];                          // -> ds_load_b64
        bfr.y = bsh[lr][ka + 1];
        c = __builtin_amdgcn_wmma_f32_16x16x4_f32(
                /*neg_a=*/false, a, /*neg_b=*/false, bfr,
                /*c_mod=*/(short)0, c, /*reuse_a=*/false, /*reuse_b=*/false);
    }

    // fused log epilogue + store: out[(b*998+f)*80 + mel]
#pragma unroll
    for (int r = 0; r < 8; ++r) {
        const long frameRow = (long)mTile * 16 + r + 8 * half;
        out[frameRow * NMEL + nCol] = __logf(c[r] + 1e-8f);
    }

    // olens tail: 32 int32 values of 998, bit-cast into the float buffer
    if (blockIdx.x == 0 && nTile == 0 && wave == 0) {
        int* olens = (int*)(out + (long)ROWS * NMEL);
        olens[lane] = NF;
    }
}

// =====================================================================
extern "C" void kernel_launch(void* const* d_in, const int* in_sizes, int n_in,
                              void* d_out, int out_size, void* d_ws, size_t ws_size,
                              hipStream_t stream) {
    (void)in_sizes; (void)n_in; (void)out_size; (void)ws_size;
    const float* feat   = (const float*)d_in[0];   // (32, 160000)
    const float* window = (const float*)d_in[1];   // (400,)
    const float* mel_fb = (const float*)d_in[2];   // (80, 257)
    float* out = (float*)d_out;
    float* ws  = (float*)d_ws;

    // workspace layout (floats): aWS[ROWS*17] | gdWS[ROWS*260] | melT[260*80]
    float* aWS  = ws;
    const long aFloats  = ((long)ROWS * 17 + 255) & ~255L;
    float* gdWS = ws + aFloats;
    const long gdFloats = (long)ROWS * KPAD;
    float* melT = gdWS + gdFloats;

    mel_pack<<<dim3((KPAD * NMEL + 255) / 256), 256, 0, stream>>>(mel_fb, melT);

    dim3 g1((NF + 7) / 8, NB);          // 8 waves/block, one frame per wave
    gd_autocorr_levinson<<<g1, 256, 0, stream>>>(feat, window, aWS);

    dim3 g2(NF, NB);                    // one block per frame
    gd_phase<<<g2, 256, 0, stream>>>(aWS, gdWS);

    dim3 g3(MTILES / WPB, NTILES);      // 499 x 5 blocks, 4 tiles per block
    gd_mel_wmma<<<g3, 128, 0, stream>>>(gdWS, melT, out);
}